// CGDetector_7232724927250
// MI455X (gfx1250) — compile-verified
//
#include <hip/hip_runtime.h>

// Batched CG: one workgroup per batch element.
// A (128x128 f32, 64KB) is staged into LDS ONCE via async-to-LDS DMA, then all
// 32 CG iterations run LDS-resident. Matvec A*d uses V_WMMA_F32_16X16X4_F32
// (f32-exact; compute is nowhere near the matrix-op ceiling, so lower precision
// buys nothing here -- this kernel is HBM-once + LDS-bandwidth bound).
// Each wave owns 32 output rows = two 16-row WMMA tiles, processed in a single
// k-loop with two accumulators so the B-fragment (d) LDS loads are shared and
// the WMMA accumulate dependency chain is split in two.

#define NDIM 128
#define LDA  132   // 132 mod 64 == 4: each of the 16 rows touched by one WMMA
                   // A-fragment b64 load owns its own 4-bank group -> no LDS
                   // bank conflicts on the per-k-step ds_load_b64.
#define TPB  128   // 4 wave32 waves; wave w computes output rows 32w..32w+31

typedef float v2f __attribute__((ext_vector_type(2)));
typedef float v8f __attribute__((ext_vector_type(8)));
// b128 payload type expected by the async-to-LDS builtin (GCC vector spelling
// to exactly match the builtin's parameter pointee type).
typedef int b128_t __attribute__((vector_size(16)));

#if defined(__has_builtin)
#  if __has_builtin(__builtin_amdgcn_global_load_async_to_lds_b128)
#    define HAVE_ASYNC_LDS 1
#  endif
#  if __has_builtin(__builtin_amdgcn_wmma_f32_16x16x4_f32)
#    define HAVE_WMMA_F32 1
#  endif
#endif

__device__ __forceinline__ float block_reduce(float v, float* red) {
  // wave32 butterfly reduce, then combine 4 wave partials through LDS
  #pragma unroll
  for (int off = 16; off > 0; off >>= 1)
    v += __shfl_xor(v, off, 32);
  const int wave = threadIdx.x >> 5;
  if ((threadIdx.x & 31) == 0) red[wave] = v;
  __syncthreads();
  v = red[0] + red[1] + red[2] + red[3];
  __syncthreads();
  return v;
}

__global__ void __launch_bounds__(TPB)
cg_detector_kernel(const float* __restrict__ A,
                   const float* __restrict__ b,
                   const int*   __restrict__ iter_p,
                   float*       __restrict__ out) {
  __shared__ float A_sh[NDIM * LDA];
  __shared__ float d_sh[NDIM];
  __shared__ float Ad_sh[NDIM];
  __shared__ float red_sh[4];

  const int tid  = threadIdx.x;
  const int lane = tid & 31;
  const int wave = tid >> 5;
  const float* Ab = A + (size_t)blockIdx.x * (size_t)(NDIM * NDIM);

  // ---- stage this batch's A tile (64 KB) into LDS, once ----
#if HAVE_ASYNC_LDS
  #pragma unroll 4
  for (int i = 0; i < (NDIM * NDIM) / (4 * TPB); ++i) {   // 32 x b128 per thread
    const int c    = i * TPB + tid;    // 16-byte chunk id (4096 total)
    const int row  = c >> 5;           // 32 chunks per 128-float row
    const int col4 = (c & 31) << 2;    // float index within row
    const float* g = Ab + row * NDIM + col4;
    float*       l = &A_sh[row * LDA + col4];
    __builtin_amdgcn_global_load_async_to_lds_b128(
        (__attribute__((address_space(1))) b128_t*)g,
        (__attribute__((address_space(3))) b128_t*)l,
        /*offset=*/0, /*cpol=*/0);
  }
#else
  #pragma unroll 4
  for (int i = 0; i < (NDIM * NDIM) / (4 * TPB); ++i) {
    const int c    = i * TPB + tid;
    const int row  = c >> 5;
    const int col4 = (c & 31) << 2;
    float4 v = *(const float4*)(Ab + row * NDIM + col4);
    *(float4*)&A_sh[row * LDA + col4] = v;
  }
#endif

  // CG state: thread t owns vector element t (kept in registers)
  const float bt = b[(size_t)blockIdx.x * NDIM + tid];
  float s = 0.0f;
  float r = -bt;     // r = A*0 - b
  float d = bt;      // d = -r

#if HAVE_ASYNC_LDS
#  if defined(__has_builtin) && __has_builtin(__builtin_amdgcn_s_wait_asynccnt)
  __builtin_amdgcn_s_wait_asynccnt(0);
#  else
  asm volatile("s_wait_asynccnt 0" ::: "memory");
#  endif
#endif
  __syncthreads();

  float rr = block_reduce(r * r, red_sh);

  const int   n_iter = iter_p[0];
  const int   laneM  = lane & 15;
  const int   khi    = (lane & 16) ? 2 : 0;   // K sub-pair held by this half-wave

  for (int it = 0; it < n_iter; ++it) {
    d_sh[tid] = d;
    __syncthreads();

    // ---- Ad = A * d via V_WMMA_F32_16X16X4_F32 ----
    // B operand broadcasts d[k] to all 16 columns, so every column of the
    // 16x16 accumulator equals the matvec; lanes 0/16 extract rows 0-7 / 8-15.
    // Both of this wave's row tiles share one B-fragment load per k-step and
    // use independent accumulators (splits the WMMA RAW chain in two).
#if HAVE_WMMA_F32
    {
      const float* arow0 = &A_sh[(wave * 32 + laneM) * LDA + khi];
      const float* arow1 = arow0 + 16 * LDA;
      v8f acc0 = {0.f, 0.f, 0.f, 0.f, 0.f, 0.f, 0.f, 0.f};
      v8f acc1 = {0.f, 0.f, 0.f, 0.f, 0.f, 0.f, 0.f, 0.f};
      #pragma unroll 8
      for (int k0 = 0; k0 < NDIM; k0 += 4) {
        v2f bv = {d_sh[k0 + khi], d_sh[k0 + khi + 1]};     // broadcast d across N
        v2f a0 = {arow0[k0], arow0[k0 + 1]};               // A[m][k..k+1] per lane
        v2f a1 = {arow1[k0], arow1[k0 + 1]};
        acc0 = __builtin_amdgcn_wmma_f32_16x16x4_f32(
            false, a0, false, bv, (short)0, acc0, false, false);
        acc1 = __builtin_amdgcn_wmma_f32_16x16x4_f32(
            false, a1, false, bv, (short)0, acc1, false, false);
      }
      const int rb0 = wave * 32;        // rows of tile 0 (acc0)
      const int rb1 = wave * 32 + 16;   // rows of tile 1 (acc1)
      if (lane == 0) {
        #pragma unroll
        for (int i = 0; i < 8; ++i) { Ad_sh[rb0 + i] = acc0[i]; Ad_sh[rb1 + i] = acc1[i]; }
      } else if (lane == 16) {
        #pragma unroll
        for (int i = 0; i < 8; ++i) { Ad_sh[rb0 + 8 + i] = acc0[i]; Ad_sh[rb1 + 8 + i] = acc1[i]; }
      }
    }
#else
    {
      float accv = 0.0f;
      const float* arow = &A_sh[tid * LDA];
      #pragma unroll 8
      for (int k = 0; k < NDIM; ++k) accv += arow[k] * d_sh[k];
      Ad_sh[tid] = accv;
    }
#endif
    __syncthreads();

    const float Ad    = Ad_sh[tid];
    const float dAd   = block_reduce(d * Ad, red_sh);
    const float alpha = rr / dAd;
    s += alpha * d;
    float rn = r + alpha * Ad;
    rn = (fabsf(rn) > 1e-10f) ? rn : r;      // torch.where fidelity
    const float rr_new = block_reduce(rn * rn, red_sh);
    const float beta   = rr_new / rr;
    d  = -rn + beta * d;
    r  = rn;
    rr = rr_new;
  }

  out[(size_t)blockIdx.x * NDIM + tid] = s;
}

extern "C" void kernel_launch(void* const* d_in, const int* in_sizes, int n_in,
                              void* d_out, int out_size, void* d_ws, size_t ws_size,
                              hipStream_t stream) {
  (void)n_in; (void)d_ws; (void)ws_size; (void)out_size;
  const float* A     = (const float*)d_in[0];
  const float* b     = (const float*)d_in[1];
  const int*   it_p  = (const int*)d_in[2];
  float*       out   = (float*)d_out;
  const int batch = in_sizes[1] / NDIM;   // 4096
  cg_detector_kernel<<<batch, TPB, 0, stream>>>(A, b, it_p, out);
}